// GCN_14800457302609
// MI455X (gfx1250) — compile-verified
//
#include <hip/hip_runtime.h>
#include <hip/hip_bf16.h>
#include <math.h>

// Problem constants (from reference)
#define B_   8
#define N_   16384
#define F_   64
#define E_   262144
#define C_   10
#define BN_  (B_ * N_)        // 131072 rows
#define BNF_ (BN_ * F_)       // 8388608 floats
#define BE_  (B_ * E_)        // 2097152 edges
#define NF_  ((long)N_ * F_)  // 1048576
#define EPS_ 1e-5f

typedef float v8f __attribute__((ext_vector_type(8)));
typedef float v2f __attribute__((ext_vector_type(2)));

// ---------------------------------------------------------------- zero
__global__ void k_zero(float* __restrict__ p, int n) {
  int i = (blockIdx.x * blockDim.x + threadIdx.x) * 4;
  if (i + 3 < n) {
    *(float4*)(p + i) = make_float4(0.f, 0.f, 0.f, 0.f);
  } else {
    for (int j = i; j < n; ++j) p[j] = 0.f;
  }
}

// ---------------------------------------------------------------- degrees
// one thread per (b, edge): deg_out[src]++ , deg_in[dst]++
__global__ void k_deg(const int* __restrict__ src, const int* __restrict__ dst,
                      float* __restrict__ degO, float* __restrict__ degI) {
  int t = blockIdx.x * 256 + threadIdx.x;   // t in [0, BE_)
  int b = t >> 18;                          // E_ = 2^18
  int s = src[t], d = dst[t];
  atomicAdd(&degO[b * N_ + s], 1.f);
  atomicAdd(&degI[b * N_ + d], 1.f);
}

// in-place deg -> rsqrt(max(deg,1))
__global__ void k_rsqrt(float* __restrict__ p, int n) {
  int i = blockIdx.x * 256 + threadIdx.x;
  if (i < n) p[i] = rsqrtf(fmaxf(p[i], 1.f));
}

// ---------------------------------------------------------------- aggregation
// agg[dst] += feat[src] * rsqrt(deg_out[src]);  16 lanes per edge, float4 each
__global__ __launch_bounds__(256) void k_aggregate(
    const float* __restrict__ feat, const int* __restrict__ src,
    const int* __restrict__ dst, const float* __restrict__ rsO,
    float* __restrict__ agg) {
  unsigned t = blockIdx.x * 256u + threadIdx.x;
  int eg = (int)(t >> 4);          // global edge index (b*E + e)
  int l  = (int)(t & 15u);         // float4 lane within the 64-float row
  int b  = eg >> 18;
  int s  = src[eg], d = dst[eg];
  float sc = rsO[b * N_ + s];
  float4 v = ((const float4*)feat)[(b * N_ + s) * 16 + l];
  float* o = agg + ((b * N_ + d) * 64 + l * 4);
  atomicAdd(o + 0, v.x * sc);
  atomicAdd(o + 1, v.y * sc);
  atomicAdd(o + 2, v.z * sc);
  atomicAdd(o + 3, v.w * sc);
}

// ---------------------------------------------------------------- WMMA GEMM
// out[m, :] = act( (agg[m, :] * rsI[m]) @ W + bias )
// M = BN_ rows, K = 64, N = 64. 256 threads = 8 waves, 128 rows / WG.
// Each wave: 16 rows x 64 cols using V_WMMA_F32_16X16X4_F32, 16 K-steps x 4 N-tiles.
template <bool RELU>
__global__ __launch_bounds__(256) void k_gemm(
    const float* __restrict__ agg, const float* __restrict__ rsI,
    const float* __restrict__ W, const float* __restrict__ bias,
    float* __restrict__ out) {
  // LDS: padded stride 68 to spread the 64-bank layout (row stride 272B -> 4-bank shift/row)
  __shared__ float sW[64 * 68];
  __shared__ float sA[128 * 68];
  __shared__ float sb[64];

  const int tid = threadIdx.x;
  const int rowBase = blockIdx.x * 128;

  // stage W (64x64) into LDS
  for (int i = tid; i < 64 * 16; i += 256) {
    int k = i >> 4, c4 = (i & 15) * 4;
    float4 w4 = ((const float4*)W)[i];
    float* p = &sW[k * 68 + c4];
    p[0] = w4.x; p[1] = w4.y; p[2] = w4.z; p[3] = w4.w;
  }
  if (tid < 64) sb[tid] = bias[tid];

  // stage 128 rows of A, pre-scaled by rsqrt(deg_in)
  for (int i = tid; i < 128 * 16; i += 256) {
    int r = i >> 4, c4 = (i & 15) * 4;
    int gr = rowBase + r;
    float s = rsI[gr];
    float4 a4 = ((const float4*)agg)[gr * 16 + (i & 15)];
    float* p = &sA[r * 68 + c4];
    p[0] = a4.x * s; p[1] = a4.y * s; p[2] = a4.z * s; p[3] = a4.w * s;
  }
  __syncthreads();

  const int wave = tid >> 5;
  const int lane = tid & 31;
  const int mr   = lane & 15;   // row-within-tile (A) / col-within-tile (B,D)
  const int hi   = lane >> 4;   // K-pair select per ISA 16x4 f32 layout
  const int aRow = wave * 16 + mr;

  v8f acc0 = {}, acc1 = {}, acc2 = {}, acc3 = {};

#pragma unroll
  for (int k0 = 0; k0 < 64; k0 += 4) {
    const int koff = k0 + 2 * hi;  // VGPR0: K=k0(+2*hi), VGPR1: +1
    v2f a;
    a.x = sA[aRow * 68 + koff];
    a.y = sA[aRow * 68 + koff + 1];
    v2f b0, b1, b2, b3;
    const float* w0 = &sW[koff * 68];
    const float* w1 = &sW[(koff + 1) * 68];
    b0.x = w0[ 0 + mr]; b0.y = w1[ 0 + mr];
    b1.x = w0[16 + mr]; b1.y = w1[16 + mr];
    b2.x = w0[32 + mr]; b2.y = w1[32 + mr];
    b3.x = w0[48 + mr]; b3.y = w1[48 + mr];
    acc0 = __builtin_amdgcn_wmma_f32_16x16x4_f32(false, a, false, b0, (short)0, acc0, false, false);
    acc1 = __builtin_amdgcn_wmma_f32_16x16x4_f32(false, a, false, b1, (short)0, acc1, false, false);
    acc2 = __builtin_amdgcn_wmma_f32_16x16x4_f32(false, a, false, b2, (short)0, acc2, false, false);
    acc3 = __builtin_amdgcn_wmma_f32_16x16x4_f32(false, a, false, b3, (short)0, acc3, false, false);
  }

  // D layout: vgpr v, lanes 0-15 -> row v, lanes 16-31 -> row v+8; col = lane&15
#pragma unroll
  for (int v = 0; v < 8; ++v) {
    int row = rowBase + wave * 16 + v + 8 * hi;
    float r0 = acc0[v] + sb[ 0 + mr];
    float r1 = acc1[v] + sb[16 + mr];
    float r2 = acc2[v] + sb[32 + mr];
    float r3 = acc3[v] + sb[48 + mr];
    if (RELU) {
      r0 = fmaxf(r0, 0.f); r1 = fmaxf(r1, 0.f);
      r2 = fmaxf(r2, 0.f); r3 = fmaxf(r3, 0.f);
    }
    float* o = out + row * 64;
    o[ 0 + mr] = r0; o[16 + mr] = r1; o[32 + mr] = r2; o[48 + mr] = r3;
  }
}

// ---------------------------------------------------------------- BN stats
// per-channel sum / sumsq over all BN_ rows; 128 rows per block
__global__ __launch_bounds__(256) void k_bnstats(const float* __restrict__ h,
                                                 float* __restrict__ stats) {
  __shared__ float sS[4][64], sQ[4][64];
  int tid = threadIdx.x;
  int c = tid & 63, g = tid >> 6;
  int base = blockIdx.x * 128;
  float s = 0.f, q = 0.f;
  for (int j = 0; j < 32; ++j) {
    int r = base + g + 4 * j;
    float v = h[r * 64 + c];
    s += v; q += v * v;
  }
  sS[g][c] = s; sQ[g][c] = q;
  __syncthreads();
  if (g == 0) {
    s = sS[0][c] + sS[1][c] + sS[2][c] + sS[3][c];
    q = sQ[0][c] + sQ[1][c] + sQ[2][c] + sQ[3][c];
    atomicAdd(&stats[c], s);
    atomicAdd(&stats[64 + c], q);
  }
}

// stats -> per-channel affine:  scale = gamma*rsqrt(var+eps), shift = beta - mean*scale
__global__ void k_bnfinal(float* __restrict__ stats, const float* __restrict__ gamma,
                          const float* __restrict__ beta) {
  int c = threadIdx.x;  // 64 threads
  const float inv = 1.f / (float)BN_;
  float mean = stats[c] * inv;
  float var  = stats[64 + c] * inv - mean * mean;
  float scale = gamma[c] * rsqrtf(var + EPS_);
  float shift = beta[c] - mean * scale;
  stats[c] = scale;
  stats[64 + c] = shift;
}

// ---------------------------------------------------------------- output init
__global__ void k_initout(float* __restrict__ out, const float* __restrict__ lin_b) {
  int i = threadIdx.x;
  if (i < B_ * C_) out[i] = lin_b[i % C_];
}

// ---------------------------------------------------------------- fused BN + final linear
// out[b,c] += sum_i (h[b,i]*scale[f] + shift[f]) * lin_W[c,i],  f = i % 64
#define LIN_BLOCKS 64
__global__ __launch_bounds__(256) void k_linear(const float* __restrict__ h,
                                                const float* __restrict__ lin_W,
                                                const float* __restrict__ stats,
                                                float* __restrict__ out) {
  __shared__ float sScale[64], sShift[64];
  int tid = threadIdx.x;
  if (tid < 64) { sScale[tid] = stats[tid]; sShift[tid] = stats[64 + tid]; }
  __syncthreads();

  const long chunk = NF_ / LIN_BLOCKS;     // 16384
  const long base  = (long)blockIdx.x * chunk;

  float acc[B_][C_];
#pragma unroll
  for (int b = 0; b < B_; ++b)
#pragma unroll
    for (int c = 0; c < C_; ++c) acc[b][c] = 0.f;

  for (long i = base + tid; i < base + chunk; i += 256) {
    int f = (int)(i & 63);
    float sc = sScale[f], sh = sShift[f];
    __builtin_prefetch(lin_W + i + 8192, 0, 0);  // stream hint -> global_prefetch_b8
    float hv[B_];
#pragma unroll
    for (int b = 0; b < B_; ++b) hv[b] = h[(long)b * NF_ + i] * sc + sh;
#pragma unroll
    for (int c = 0; c < C_; ++c) {
      float w = lin_W[(long)c * NF_ + i];
#pragma unroll
      for (int b = 0; b < B_; ++b) acc[b][c] = fmaf(hv[b], w, acc[b][c]);
    }
  }

  int lane = tid & 31;
#pragma unroll
  for (int b = 0; b < B_; ++b)
#pragma unroll
    for (int c = 0; c < C_; ++c) {
      float v = acc[b][c];
      for (int off = 16; off > 0; off >>= 1) v += __shfl_down(v, off, 32);
      if (lane == 0) atomicAdd(&out[b * C_ + c], v);
    }
}

// ---------------------------------------------------------------- launch
extern "C" void kernel_launch(void* const* d_in, const int* in_sizes, int n_in,
                              void* d_out, int out_size, void* d_ws, size_t ws_size,
                              hipStream_t stream) {
  (void)in_sizes; (void)n_in; (void)out_size; (void)ws_size;
  const float* x     = (const float*)d_in[0];
  const int*   esrc  = (const int*)d_in[1];
  const int*   edst  = (const int*)d_in[2];
  const float* W1    = (const float*)d_in[3];
  const float* b1    = (const float*)d_in[4];
  const float* W2    = (const float*)d_in[5];
  const float* b2    = (const float*)d_in[6];
  const float* gamma = (const float*)d_in[7];
  const float* beta  = (const float*)d_in[8];
  const float* linW  = (const float*)d_in[9];
  const float* linb  = (const float*)d_in[10];
  float* out = (float*)d_out;

  // workspace layout (~68.2 MB)
  char* ws = (char*)d_ws;
  float* bufA  = (float*)ws;                          // BNF_ floats (agg1 / agg2)
  float* bufB  = (float*)(ws + (size_t)BNF_ * 4);     // BNF_ floats (h1 -> h2)
  float* rsO   = (float*)(ws + (size_t)BNF_ * 8);     // BN_ floats
  float* rsI   = rsO + BN_;                           // BN_ floats
  float* stats = rsI + BN_;                           // 128 floats

  dim3 blk(256);

  k_zero<<<BNF_ / 1024, blk, 0, stream>>>(bufA, BNF_);
  k_zero<<<(2 * BN_ + 128 + 1023) / 1024, blk, 0, stream>>>(rsO, 2 * BN_ + 128);
  k_deg<<<BE_ / 256, blk, 0, stream>>>(esrc, edst, rsO, rsI);
  k_rsqrt<<<(2 * BN_) / 256, blk, 0, stream>>>(rsO, 2 * BN_);

  // layer 1: aggregate, then WMMA GEMM + bias + relu
  k_aggregate<<<(BE_ * 16) / 256, blk, 0, stream>>>(x, esrc, edst, rsO, bufA);
  k_gemm<true><<<BN_ / 128, blk, 0, stream>>>(bufA, rsI, W1, b1, bufB);

  // layer 2
  k_zero<<<BNF_ / 1024, blk, 0, stream>>>(bufA, BNF_);
  k_aggregate<<<(BE_ * 16) / 256, blk, 0, stream>>>(bufB, esrc, edst, rsO, bufA);
  k_gemm<false><<<BN_ / 128, blk, 0, stream>>>(bufA, rsI, W2, b2, bufB);

  // batchnorm + fused final linear
  k_bnstats<<<BN_ / 128, blk, 0, stream>>>(bufB, stats);
  k_bnfinal<<<1, 64, 0, stream>>>(stats, gamma, beta);
  k_initout<<<1, 128, 0, stream>>>(out, linb);
  k_linear<<<LIN_BLOCKS, blk, 0, stream>>>(bufB, linW, stats, out);
}